// YoloLoss_15831249453726
// MI455X (gfx1250) — compile-verified
//
#include <hip/hip_runtime.h>

// YOLO loss, MI455X (gfx1250, wave32).
// Bandwidth-bound: ~96MB of predictions @23.3TB/s => ~4.2us floor.
// Strategy: scatter boxes->cells (atomicMax = last-wins), then one thread per
// grid cell. Tile staging uses gfx1250 async-to-LDS DMA
// (global_load_async_to_lds_b64 + s_wait_asynccnt) instead of a VGPR round
// trip; per-wave reduction via V_WMMA_F32_16X16X4_F32 (A=ones trick) +
// ds_swizzle xor-reduce; one global_atomic_add_f32 per block.

#define S14   14
#define CGRID (S14 * S14)          // 196 cells per image
#define N_EL  30                   // 2*5 + 20 channels
#define BATCH 4096
#define NBOX  16
#define CELLS (BATCH * CGRID)      // 802816 (= 3136 blocks * 256 threads exactly)

typedef __attribute__((ext_vector_type(2))) float v2f;
typedef __attribute__((ext_vector_type(8))) float v8f;

__global__ void yolo_init(int* __restrict__ cellmap, float* __restrict__ out) {
  int i = blockIdx.x * blockDim.x + threadIdx.x;
  if (i < CELLS) cellmap[i] = 0;
  if (i == 0) out[0] = 0.0f;
}

__global__ void yolo_scatter(const float* __restrict__ boxes,
                             int* __restrict__ cellmap) {
  int idx = blockIdx.x * blockDim.x + threadIdx.x;   // one thread per (batch, box)
  if (idx >= BATCH * NBOX) return;
  int b = idx / NBOX;
  int k = idx - b * NBOX;
  const float* bx = boxes + (size_t)idx * 4;
  float cx = 0.5f * (bx[0] + bx[2]);
  float cy = 0.5f * (bx[1] + bx[3]);
  int ix = (int)ceilf(cx * 14.0f) - 1; ix = min(max(ix, 0), 13);
  int iy = (int)ceilf(cy * 14.0f) - 1; iy = min(max(iy, 0), 13);
  // JAX .at[].set with duplicate indices ~ sequential last-wins: keep max k.
  atomicMax(&cellmap[b * CGRID + iy * S14 + ix], k + 1);
}

__global__ void __launch_bounds__(256) yolo_loss(
    const float* __restrict__ pred,
    const float* __restrict__ boxes,
    const int*   __restrict__ classes,
    const int*   __restrict__ cellmap,
    float* __restrict__ out) {
  __shared__ float rows[256 * N_EL];   // 30720 B staged tile
  __shared__ float wsum[8];

  const int tid = threadIdx.x;
  const size_t base = (size_t)blockIdx.x * (256 * N_EL);

  // Speculative prefetch of the next block's tile -> global_prefetch_b8.
  __builtin_prefetch(pred + base + 256 * N_EL, 0, 0);

  // ---- gfx1250 async DMA staging: memory -> LDS, no VGPR round trip ----
  // 3840 x b64 per block (= 256 rows * 30 f32), fully coalesced, tracked by
  // ASYNCcnt. LDS byte offset comes from the addrspace(3) view of `rows`.
  {
    unsigned ldsbase =
        (unsigned)(size_t)(__attribute__((address_space(3))) char*)rows;
    const float2* __restrict__ src = (const float2*)(pred + base);
#pragma unroll
    for (int j = 0; j < 15; ++j) {
      unsigned l = ldsbase + (unsigned)((tid + j * 256) * 8);
      unsigned long long g = (unsigned long long)(src + tid + j * 256);
      asm volatile("global_load_async_to_lds_b64 %0, %1, off"
                   :: "v"(l), "v"(g) : "memory");
    }
    asm volatile("s_wait_asynccnt 0x0" ::: "memory");
  }
  __syncthreads();

  const int cell = blockIdx.x * 256 + tid;       // grid exactly covers CELLS
  const int b = cell / CGRID;
  const float* __restrict__ p = &rows[tid * N_EL];

  float loss = 0.0f;
  const int m = cellmap[cell];
  if (m == 0) {
    // no-object cells: L_NOOBJ * (p4^2 + p9^2), targets are zero
    loss = 0.5f * (p[4] * p[4] + p[9] * p[9]);
  } else {
    const int k = m - 1;
    const float* bx = boxes + ((size_t)b * NBOX + k) * 4;
    const int cls = classes[b * NBOX + k];
    const float w  = bx[2] - bx[0];
    const float h  = bx[3] - bx[1];
    const float cx = 0.5f * (bx[0] + bx[2]);
    const float cy = 0.5f * (bx[1] + bx[3]);
    int ix = (int)ceilf(cx * 14.0f) - 1; ix = min(max(ix, 0), 13);
    int iy = (int)ceilf(cy * 14.0f) - 1; iy = min(max(iy, 0), 13);
    const float dx = cx * 14.0f - (float)ix;
    const float dy = cy * 14.0f - (float)iy;

    // class loss: one-hot at channel 9+cls (cls==0 folds onto conf ch9 -> all-zero here)
    float cl = 0.0f;
#pragma unroll
    for (int c = 10; c < 30; ++c) {
      float t = (cls >= 1 && c == 9 + cls) ? 1.0f : 0.0f;
      float d = p[c] - t;
      cl += d * d;
    }

    // corner-format jaccard applied verbatim to (dx,dy,w,h) rows
    const float area_b = (w - dx) * (h - dy);
    float iou[2];
#pragma unroll
    for (int i2 = 0; i2 < 2; ++i2) {
      const float* q = p + i2 * 5;
      float ltx = fmaxf(q[0], dx), lty = fmaxf(q[1], dy);
      float rbx = fminf(q[2], w),  rby = fminf(q[3], h);
      float wx = fmaxf(rbx - ltx, 0.0f), wy = fmaxf(rby - lty, 0.0f);
      float inter = wx * wy;
      float area_a = (q[2] - q[0]) * (q[3] - q[1]);
      float un = area_a + area_b - inter;
      iou[i2] = inter / ((un == 0.0f) ? 1.0f : un);
    }
    const int r = (iou[1] > iou[0]) ? 1 : 0;       // argmax, tie -> box 0
    const float* q = p + r * 5;
    const float contain = (q[4] - 1.0f) * (q[4] - 1.0f);
    const float sw = sqrtf(q[2]) - sqrtf(w);
    const float sh = sqrtf(q[3]) - sqrtf(h);
    const float loc = (q[0] - dx) * (q[0] - dx) + (q[1] - dy) * (q[1] - dy)
                    + sw * sw + sh * sh;
    loss = 5.0f * loc + contain + cl;              // L_COORD=5, coordf=1, noobjf=0
  }

  // ---- per-wave reduction via V_WMMA_F32_16X16X4_F32 (A = ones) ----
  // Each lane's partial occupies one (K,N) slot of B (2nd B VGPR = 0), so
  // D[0][n] = column sums; summing D row 0 over a 16-lane group gives the
  // wave sum. Executed by ALL waves with full EXEC (WMMA requirement).
  v2f a; a.x = 1.0f; a.y = 1.0f;
  v2f bv; bv.x = loss; bv.y = 0.0f;
  v8f c = {};
  c = __builtin_amdgcn_wmma_f32_16x16x4_f32(
      /*neg_a=*/false, a, /*neg_b=*/false, bv,
      /*c_mod=*/(short)0, c, /*reuse_a=*/false, /*reuse_b=*/false);
  float rsum = c[0];                      // lanes 0..15 hold the 16 column sums
  // xor-reduce across the 16-lane group with single-instruction ds_swizzle
  // (group-of-32 mode: offset = xor_mask<<10 | and_mask 0x1F)
  rsum += __int_as_float(__builtin_amdgcn_ds_swizzle(__float_as_int(rsum), 0x201F)); // xor 8
  rsum += __int_as_float(__builtin_amdgcn_ds_swizzle(__float_as_int(rsum), 0x101F)); // xor 4
  rsum += __int_as_float(__builtin_amdgcn_ds_swizzle(__float_as_int(rsum), 0x081F)); // xor 2
  rsum += __int_as_float(__builtin_amdgcn_ds_swizzle(__float_as_int(rsum), 0x041F)); // xor 1

  const int lane = tid & 31, wave = tid >> 5;
  if (lane == 0) wsum[wave] = rsum;
  __syncthreads();
  if (tid == 0) {
    float s = 0.0f;
#pragma unroll
    for (int i = 0; i < 8; ++i) s += wsum[i];
    atomicAdd(out, s);                    // global_atomic_add_f32, 3136 total
  }
}

extern "C" void kernel_launch(void* const* d_in, const int* in_sizes, int n_in,
                              void* d_out, int out_size, void* d_ws, size_t ws_size,
                              hipStream_t stream) {
  const float* pred    = (const float*)d_in[0];
  const float* boxes   = (const float*)d_in[1];
  const int*   classes = (const int*)d_in[2];
  float* out = (float*)d_out;
  int* cellmap = (int*)d_ws;             // CELLS ints = 3.2 MB scratch

  const int blocks = CELLS / 256;        // 3136, exact
  yolo_init<<<blocks, 256, 0, stream>>>(cellmap, out);
  yolo_scatter<<<(BATCH * NBOX + 255) / 256, 256, 0, stream>>>(boxes, cellmap);
  yolo_loss<<<blocks, 256, 0, stream>>>(pred, boxes, classes, cellmap, out);
}